// GridsampleNorm_37641093382900
// MI455X (gfx1250) — compile-verified
//
#include <hip/hip_runtime.h>
#include <hip/hip_bf16.h>

// grid_sample bilinear, zeros padding, NCHW fp32
// N=8, C=64, H=W=256, Ho=Wo=256
// L2-gather-bound: x (134MB) is L2-resident on MI455X (192MB L2).
// Strategy: 2 output pixels / thread, b64 pair gathers for horizontal corner
// pairs, branchless validity folded into weights, NT float2 stores, prefetch.

#define GS_N  8
#define GS_C  64
#define GS_H  256
#define GS_W  256
#define GS_HW (GS_H * GS_W)          // 65536 floats per channel plane

typedef float vf2 __attribute__((ext_vector_type(2)));
typedef float vf4 __attribute__((ext_vector_type(4)));

// 8-byte load that is only guaranteed 4-byte aligned (legal under DWORD
// alignment mode: requirement is min(elem,4B)); compiler emits b64.
static __device__ __forceinline__ vf2 ldpair(const float* p) {
    vf2 r;
    __builtin_memcpy(&r, p, sizeof(r));
    return r;
}

__global__ __launch_bounds__(256) void GridsampleNorm_kernel(
    const float* __restrict__ x,     // [N, C, H, W]
    const float* __restrict__ grid,  // [N, Ho, Wo, 2]
    float* __restrict__ out)         // [N, C, Ho, Wo]
{
    const int tid = blockIdx.x * blockDim.x + threadIdx.x;   // over N*Ho*(Wo/2)
    const int wo = (tid & 127) << 1;         // even output column
    const int ho = (tid >> 7) & 255;
    const int n  = tid >> 15;

    // Prefetch grid stream ahead (gfx1250 global_prefetch_b8).
    __builtin_prefetch(reinterpret_cast<const vf4*>(grid) + tid + 1024, 0, 1);

    // Coalesced 16B grid load: two (gx,gy) pairs for pixels wo, wo+1.
    const vf4 g = reinterpret_cast<const vf4*>(grid)[tid];

    // Per-pixel sampling state.
    int   oA[2], oB[2];                      // row0/row1 pair offsets per pixel
    float wA0[2], wB0[2], wA1[2], wB1[2];    // pair-element weights per row

    #pragma unroll
    for (int k = 0; k < 2; ++k) {
        const float gx = (k == 0) ? g.x : g.z;
        const float gy = (k == 0) ? g.y : g.w;
        // ix = ((gx+1)*W - 1) * 0.5 ; W == 256 -> (gx+1)*128 - 0.5
        const float ix = (gx + 1.0f) * 128.0f - 0.5f;
        const float iy = (gy + 1.0f) * 128.0f - 0.5f;

        const float x0f = floorf(ix);
        const float y0f = floorf(iy);
        const int x0 = (int)x0f;
        const int y0 = (int)y0f;
        const int y1 = y0 + 1;

        const float tx = ix - x0f;
        const float ty = iy - y0f;
        const float colL = 1.0f - tx;
        const float colR = tx;

        // Branchless corner->pair-element remap.
        // Pair load at xL covers indices {xL, xL+1}; corner x0 sits at element
        // d0 = x0-xL, corner x1 at d0+1. A corner contributes iff its element
        // index is 0 or 1 (then it is automatically inside [0, W-1]).
        const int   xL = min(max(x0, 0), GS_W - 2);
        const int   d0 = x0 - xL;            // in {<-1, -1, 0, 1, >1}
        const int   d1 = d0 + 1;
        const float cA = (d0 == 0 ? colL : 0.0f) + (d1 == 0 ? colR : 0.0f);
        const float cB = (d0 == 1 ? colL : 0.0f) + (d1 == 1 ? colR : 0.0f);

        // Branchless row validity + clamp.
        const float vy0 = ((unsigned)y0 < (unsigned)GS_H) ? 1.0f : 0.0f;
        const float vy1 = ((unsigned)y1 < (unsigned)GS_H) ? 1.0f : 0.0f;
        const int yc0 = min(max(y0, 0), GS_H - 1);
        const int yc1 = min(max(y1, 0), GS_H - 1);

        const float r0 = (1.0f - ty) * vy0;
        const float r1 = ty * vy1;
        wA0[k] = cA * r0;  wB0[k] = cB * r0;
        wA1[k] = cA * r1;  wB1[k] = cB * r1;
        oA[k] = yc0 * GS_W + xL;
        oB[k] = yc1 * GS_W + xL;
    }

    const float* __restrict__ p = x + (size_t)n * GS_C * GS_HW;
    size_t oidx = (size_t)n * GS_C * GS_HW + (size_t)ho * GS_W + wo;

    // Per channel: 4 pair gathers (L2-resident) + 8 FMAs + 1 NT float2 store.
    #pragma unroll 8
    for (int c = 0; c < GS_C; ++c) {
        const vf2 a0 = ldpair(p + oA[0]);
        const vf2 b0 = ldpair(p + oB[0]);
        const vf2 a1 = ldpair(p + oA[1]);
        const vf2 b1 = ldpair(p + oB[1]);

        vf2 v;
        v.x = fmaf(a0.x, wA0[0], fmaf(a0.y, wB0[0],
              fmaf(b0.x, wA1[0], b0.y * wB1[0])));
        v.y = fmaf(a1.x, wA0[1], fmaf(a1.y, wB0[1],
              fmaf(b1.x, wA1[1], b1.y * wB1[1])));

        // Stream output (NT): keep x resident in the 192MB L2.
        __builtin_nontemporal_store(v, reinterpret_cast<vf2*>(out + oidx));

        p += GS_HW;
        oidx += GS_HW;
    }
}

extern "C" void kernel_launch(void* const* d_in, const int* in_sizes, int n_in,
                              void* d_out, int out_size, void* d_ws, size_t ws_size,
                              hipStream_t stream) {
    (void)in_sizes; (void)n_in; (void)out_size; (void)d_ws; (void)ws_size;
    const float* x    = (const float*)d_in[0];   // [8,64,256,256] fp32
    const float* grid = (const float*)d_in[1];   // [8,256,256,2] fp32
    float* out        = (float*)d_out;           // [8,64,256,256] fp32

    const int total = GS_N * GS_H * (GS_W / 2);  // 262144 threads, 2 pixels each
    const int block = 256;                       // 8 wave32 waves per block
    const int nblk  = total / block;             // 1024 blocks
    GridsampleNorm_kernel<<<nblk, block, 0, stream>>>(x, grid, out);
}